// SEModule_33706903339438
// MI455X (gfx1250) — compile-verified
//
#include <hip/hip_runtime.h>
#include <hip/hip_bf16.h>

typedef __attribute__((ext_vector_type(2)))  float v2f;
typedef __attribute__((ext_vector_type(8)))  float v8f;

#define CHANNELS 256
#define MID      16
#define BATCH    16
#define HW       (128 * 128)   // 16384 elements per (b,c) slice

// ---------------------------------------------------------------------------
// Kernel 1: global average pool. One block per (b,c) slice (4096 blocks).
// Slice = 16384 contiguous floats = 4096 float4 -> 16 float4 per thread.
// ---------------------------------------------------------------------------
__global__ void se_pool(const float* __restrict__ x, float* __restrict__ s) {
    const int slice = blockIdx.x;                      // b*CHANNELS + c
    const float4* xs = (const float4*)(x + (size_t)slice * HW);
    const int t = threadIdx.x;

    float sum = 0.f;
#pragma unroll
    for (int j = 0; j < 16; ++j) {
        float4 v = xs[t + 256 * j];                    // coalesced b128 loads
        sum += (v.x + v.y) + (v.z + v.w);
    }

    __shared__ float red[256];
    red[t] = sum;
    __syncthreads();
#pragma unroll
    for (int off = 128; off > 0; off >>= 1) {
        if (t < off) red[t] += red[t + off];
        __syncthreads();
    }
    if (t == 0) s[slice] = red[0] * (1.0f / (float)HW);
}

// ---------------------------------------------------------------------------
// Kernel 2: the two tiny FC layers on ONE wave32 using v_wmma_f32_16x16x4_f32.
//   h[b,m] = relu( sum_c s[b,c]*w1[m,c] + b1[m] )      (M=16, N=16, K=256)
//   g[b,c] = sigmoid( sum_m h[b,m]*w2[c,m] + b2[c] )   (M=16, N=256, K=16)
// A layout (16x4 f32, MxK): lanes 0-15 hold M=lane {K=kb+0,kb+1} in VGPR0/1,
// lanes 16-31 hold M=lane-16 {K=kb+2,kb+3}.  B (4x16, KxN) mirrors with N=lane%16.
// C/D layout: VGPR v, lanes 0-15 -> (M=v, N=lane); lanes 16-31 -> (M=v+8, N=lane-16).
// ---------------------------------------------------------------------------
__global__ void se_gate(const float* __restrict__ s,    // [16,256]
                        const float* __restrict__ w1,   // [16,256]  (m,c)
                        const float* __restrict__ b1,   // [16]
                        const float* __restrict__ w2,   // [256,16]  (c,m)
                        const float* __restrict__ b2,   // [256]
                        float* __restrict__ g)          // [16,256]
{
    const int lane  = threadIdx.x;     // 0..31
    const int row   = lane & 15;       // M index (A) / N index (B) within half
    const int hi    = lane >> 4;       // 0: lanes 0-15, 1: lanes 16-31
    const int kOff  = hi ? 2 : 0;
    const int rbase = hi ? 8 : 0;

#if __has_builtin(__builtin_amdgcn_wmma_f32_16x16x4_f32)
    // ---- GEMM 1: h = s * w1^T  (K=256 in 64 steps of 4) ----
    v8f acc = {};
#pragma unroll 8
    for (int kb = 0; kb < CHANNELS; kb += 4) {
        v2f a, b;
        a.x = s [row * CHANNELS + kb + kOff + 0];
        a.y = s [row * CHANNELS + kb + kOff + 1];
        b.x = w1[row * CHANNELS + kb + kOff + 0];   // B[k,n] = w1[n,k]
        b.y = w1[row * CHANNELS + kb + kOff + 1];
        acc = __builtin_amdgcn_wmma_f32_16x16x4_f32(
            /*neg_a=*/false, a, /*neg_b=*/false, b,
            /*c_mod=*/(short)0, acc, /*reuse_a=*/false, /*reuse_b=*/false);
    }

    // bias + relu; stash h (C/D layout -> plain [M][N] array in LDS)
    __shared__ float hbuf[16][17];
    {
        const float bias1 = b1[row];                 // bias indexed by N=row
#pragma unroll
        for (int v = 0; v < 8; ++v) {
            float hv = acc[v] + bias1;
            hbuf[rbase + v][row] = hv > 0.f ? hv : 0.f;
        }
    }
    __syncthreads();

    // ---- GEMM 2: g_pre = h * w2^T  (N=256 in 16 tiles, K=16 in 4 steps) ----
    v2f a2[4];
#pragma unroll
    for (int kk = 0; kk < 4; ++kk) {                 // A2[M=row, K] = hbuf[row][K]
        const int kb = kk * 4 + kOff;
        a2[kk].x = hbuf[row][kb + 0];
        a2[kk].y = hbuf[row][kb + 1];
    }

    for (int t = 0; t < 16; ++t) {                   // N tile: channels 16t..16t+15
        v8f accg = {};
        const int c = t * 16 + row;                  // this lane's B column channel
#pragma unroll
        for (int kk = 0; kk < 4; ++kk) {
            const int kb = kk * 4 + kOff;
            v2f bf;
            bf.x = w2[c * MID + kb + 0];             // B2[k,n] = w2[n_channel, k]
            bf.y = w2[c * MID + kb + 1];
            accg = __builtin_amdgcn_wmma_f32_16x16x4_f32(
                false, a2[kk], false, bf, (short)0, accg, false, false);
        }
        const float bias2 = b2[c];
#pragma unroll
        for (int v = 0; v < 8; ++v) {
            float pre = accg[v] + bias2;
            g[(size_t)(rbase + v) * CHANNELS + c] = 1.0f / (1.0f + __expf(-pre));
        }
    }
#else
    // Scalar fallback (only if the f32 x4 WMMA builtin is unavailable).
    __shared__ float hbuf[16][16];
    for (int i = lane; i < BATCH * MID; i += 32) {
        const int b = i >> 4, m = i & 15;
        float a0 = b1[m];
        for (int c = 0; c < CHANNELS; ++c) a0 += s[b * CHANNELS + c] * w1[m * CHANNELS + c];
        hbuf[b][m] = a0 > 0.f ? a0 : 0.f;
    }
    __syncthreads();
    for (int i = lane; i < BATCH * CHANNELS; i += 32) {
        const int b = i >> 8, c = i & 255;
        float a0 = b2[c];
        for (int m = 0; m < MID; ++m) a0 += hbuf[b][m] * w2[c * MID + m];
        g[i] = 1.0f / (1.0f + __expf(-a0));
    }
#endif
}

// ---------------------------------------------------------------------------
// Kernel 3: out = x * g[b,c].  16 blocks per slice -> g lookup is block-uniform
// (scalar load + broadcast).  One float4 load + float4 store per thread.
// ---------------------------------------------------------------------------
__global__ void se_scale(const float* __restrict__ x,
                         const float* __restrict__ g,
                         float* __restrict__ out) {
    const int slice = blockIdx.x >> 4;               // 4096 float4/slice, 256/block
    const float gv = g[slice];                       // uniform -> s_load
    const size_t i4 = (size_t)blockIdx.x * 256 + threadIdx.x;
    float4 v = ((const float4*)x)[i4];
    v.x *= gv; v.y *= gv; v.z *= gv; v.w *= gv;
    ((float4*)out)[i4] = v;
}

// ---------------------------------------------------------------------------
extern "C" void kernel_launch(void* const* d_in, const int* in_sizes, int n_in,
                              void* d_out, int out_size, void* d_ws, size_t ws_size,
                              hipStream_t stream) {
    const float* x  = (const float*)d_in[0];   // [16,256,128,128]
    const float* w1 = (const float*)d_in[1];   // [16,256]
    const float* b1 = (const float*)d_in[2];   // [16]
    const float* w2 = (const float*)d_in[3];   // [256,16]
    const float* b2 = (const float*)d_in[4];   // [256]
    float* out = (float*)d_out;

    float* s = (float*)d_ws;                   // [16*256] pooled means
    float* g = s + BATCH * CHANNELS;           // [16*256] gates

    se_pool <<<BATCH * CHANNELS, 256, 0, stream>>>(x, s);
    se_gate <<<1, 32, 0, stream>>>(s, w1, b1, w2, b2, g);
    se_scale<<<(BATCH * CHANNELS * HW) / (4 * 256), 256, 0, stream>>>(x, g, out);
}